// Masker_35682588295617
// MI455X (gfx1250) — compile-verified
//
#include <hip/hip_runtime.h>
#include <hip/hip_bf16.h>
#include <stdint.h>

#define N_B 8
#define L_PTS 16384
#define D_DIM 256
#define K_CENT 25
#define LEN_KEEP 4096
#define LEN_MASK (L_PTS - LEN_KEEP)

typedef float v2f __attribute__((ext_vector_type(2)));
typedef float v8f __attribute__((ext_vector_type(8)));

// pack (value, index) for argmax with lowest-index tie-break (values are >= 0)
__device__ __forceinline__ unsigned long long packMax(float v, int idx) {
    unsigned int b = __float_as_uint(v);
    return ((unsigned long long)b << 32) |
           (unsigned long long)(0xFFFFFFFFu - (unsigned int)idx);
}

// ---------------------------------------------------------------------------
// Kernel 1: furthest point sampling. One 1024-thread block per batch.
// Each thread keeps 16 points + running distances in registers.
// ---------------------------------------------------------------------------
__global__ void __launch_bounds__(1024)
fps_kernel(const float* __restrict__ pts, const int* __restrict__ finit,
           float* __restrict__ cent /* [N][K][3] */) {
    const int b = blockIdx.x;
    const int t = threadIdx.x;
    const float* P = pts + (size_t)b * L_PTS * 3;

    __shared__ float cbuf[3];
    __shared__ int fshare;
    __shared__ unsigned long long redbuf[32];

    float px[16], py[16], pz[16], dist[16];
#pragma unroll
    for (int i = 0; i < 16; ++i) {
        int p = i * 1024 + t;
        px[i] = P[p * 3 + 0];
        py[i] = P[p * 3 + 1];
        pz[i] = P[p * 3 + 2];
        dist[i] = 1e10f;
    }
    if (t == 0) fshare = finit[b];
    __syncthreads();

    for (int s = 0; s < K_CENT; ++s) {
        if (t == 0) {
            int f = fshare;
            float cx = P[f * 3 + 0], cy = P[f * 3 + 1], cz = P[f * 3 + 2];
            cbuf[0] = cx; cbuf[1] = cy; cbuf[2] = cz;
            float* C = cent + ((size_t)b * K_CENT + s) * 3;
            C[0] = cx; C[1] = cy; C[2] = cz;
        }
        __syncthreads();
        float cx = cbuf[0], cy = cbuf[1], cz = cbuf[2];

        unsigned long long best = 0ull;
#pragma unroll
        for (int i = 0; i < 16; ++i) {
            float dx = px[i] - cx, dy = py[i] - cy, dz = pz[i] - cz;
            float d = dx * dx + dy * dy + dz * dz;
            dist[i] = fminf(dist[i], d);
            unsigned long long k = packMax(dist[i], i * 1024 + t);
            best = (k > best) ? k : best;
        }
        // wave32 reduce
#pragma unroll
        for (int m = 16; m >= 1; m >>= 1) {
            unsigned long long o = __shfl_xor(best, m, 32);
            best = (o > best) ? o : best;
        }
        if ((t & 31) == 0) redbuf[t >> 5] = best;
        __syncthreads();
        if (t < 32) {
            best = redbuf[t];
#pragma unroll
            for (int m = 16; m >= 1; m >>= 1) {
                unsigned long long o = __shfl_xor(best, m, 32);
                best = (o > best) ? o : best;
            }
            if (t == 0)
                fshare = (int)(0xFFFFFFFFu - (unsigned int)(best & 0xFFFFFFFFull));
        }
        __syncthreads();
    }
}

// ---------------------------------------------------------------------------
// Kernel 2: dmin[b][i] = min_j ||p_i - c_j||^2  via WMMA f32 16x16x4.
// One wave = one 16-point tile. Two WMMA tiles cover 25 (padded to 32)
// centroids; padded columns carry cc=+inf so they never win the min.
// ---------------------------------------------------------------------------
__global__ void __launch_bounds__(256)
dmin_kernel(const float* __restrict__ pts, const float* __restrict__ cent,
            float* __restrict__ dmin) {
    const int b    = blockIdx.y;
    const int wave = threadIdx.x >> 5;
    const int lane = threadIdx.x & 31;
    const int half = lane >> 4;   // 0: K0/K1 rows, 1: K2/K3 rows
    const int lm   = lane & 15;
    const int m0   = (blockIdx.x * 8 + wave) * 16;

    __shared__ float cxs[32], cys[32], czs[32], ccs[32];
    __shared__ float pps[8][16];

    if (threadIdx.x < 32) {
        int j = threadIdx.x;
        float cx = 0.f, cy = 0.f, cz = 0.f, cc;
        if (j < K_CENT) {
            const float* C = cent + ((size_t)b * K_CENT + j) * 3;
            cx = C[0]; cy = C[1]; cz = C[2];
            cc = cx * cx + cy * cy + cz * cz;
        } else {
            cc = 3.0e38f;  // kill padded columns
        }
        cxs[j] = cx; cys[j] = cy; czs[j] = cz; ccs[j] = cc;
    }

    const float* P = pts + ((size_t)b * L_PTS + m0) * 3;
    float x = P[lm * 3 + 0], y = P[lm * 3 + 1], z = P[lm * 3 + 2];
    if (half == 0) pps[wave][lm] = x * x + y * y + z * z;
    __syncthreads();

    // A 16x4 f32 layout: lanes 0-15 -> {K0,K1}={x,y}; lanes 16-31 -> {K2,K3}={z,0}
    v2f a;
    a.x = half ? z : x;
    a.y = half ? 0.f : y;
    v2f b1, b2;
    b1.x = half ? czs[lm] : cxs[lm];
    b1.y = half ? 0.f : cys[lm];
    b2.x = half ? czs[16 + lm] : cxs[16 + lm];
    b2.y = half ? 0.f : cys[16 + lm];

#if __has_builtin(__builtin_amdgcn_wmma_f32_16x16x4_f32)
    v8f z8 = {};
    v8f d1 = __builtin_amdgcn_wmma_f32_16x16x4_f32(false, a, false, b1,
                                                   (short)0, z8, false, false);
    v8f d2 = __builtin_amdgcn_wmma_f32_16x16x4_f32(false, a, false, b2,
                                                   (short)0, z8, false, false);
#else
    // host-pass / fallback: scalar Gram tiles (same math)
    __shared__ float axs[8][16], ays[8][16], azs[8][16];
    if (half == 0) { axs[wave][lm] = x; ays[wave][lm] = y; azs[wave][lm] = z; }
    __syncthreads();
    v8f d1, d2;
#pragma unroll
    for (int r = 0; r < 8; ++r) {
        int M = r + half * 8;
        d1[r] = axs[wave][M] * cxs[lm] + ays[wave][M] * cys[lm] + azs[wave][M] * czs[lm];
        d2[r] = axs[wave][M] * cxs[16 + lm] + ays[wave][M] * cys[16 + lm] +
                azs[wave][M] * czs[16 + lm];
    }
#endif

    float c1 = ccs[lm], c2 = ccs[16 + lm];
#pragma unroll
    for (int r = 0; r < 8; ++r) {
        int M = r + half * 8;           // row of C/D tile held by this lane
        float pp = pps[wave][M];
        float v = fminf(pp + c1 - 2.0f * d1[r], pp + c2 - 2.0f * d2[r]);
#pragma unroll
        for (int m = 8; m >= 1; m >>= 1)
            v = fminf(v, __shfl_xor(v, m, 16));  // min over N within 16-lane half
        if (lm == 0)
            dmin[(size_t)b * L_PTS + m0 + M] = fmaxf(v, 0.0f);
    }
}

// ---------------------------------------------------------------------------
// Kernel 3: per-batch stable argsort (16384 keys) via in-LDS bitonic sort of
// packed u64 (key_bits<<32 | index). The 64KB key slab is DMA'd into LDS by
// the Tensor Data Mover (one scalar-issued tensor_load_to_lds per block,
// waited with s_wait_tensorcnt) when the TDM builtin is available.
// Epilogue scatters ids_restore/mask and gathers the small xyz/rgb outputs.
// ---------------------------------------------------------------------------
#if __has_builtin(__builtin_amdgcn_tensor_load_to_lds) && \
    __has_builtin(__builtin_amdgcn_s_wait_tensorcnt)
#define USE_TDM_LOAD 1
typedef unsigned int tdm_v4u __attribute__((ext_vector_type(4)));
typedef int          tdm_v8i __attribute__((ext_vector_type(8)));
typedef int          tdm_v4i __attribute__((ext_vector_type(4)));
#endif

__global__ void __launch_bounds__(1024)
sort_kernel(const float* __restrict__ dmin, const float* __restrict__ pts,
            const float* __restrict__ rgb, int* __restrict__ idxout,
            float* __restrict__ ids_restore, float* __restrict__ mask,
            float* __restrict__ pkeep, float* __restrict__ pmask,
            float* __restrict__ rkeep, float* __restrict__ rmask) {
    extern __shared__ unsigned long long skey[];  // 16384 * 8B = 128 KB
    const int b = blockIdx.x;
    const int t = threadIdx.x;

#ifdef USE_TDM_LOAD
    __shared__ float stage[L_PTS];  // 64 KB TDM landing zone
    if ((t >> 5) == 0) {  // one wave issues the DMA (wave-uniform branch)
        unsigned int lds_off = (unsigned int)(size_t)(void*)stage;
        unsigned long long ga =
            (unsigned long long)(size_t)(dmin + (size_t)b * L_PTS);
        // D# group 0: count=1 | lds_addr | global_addr | type=2 ("image")
        tdm_v4u g0;
        g0.x = 0x1u;
        g0.y = lds_off;
        g0.z = (unsigned int)(ga & 0xFFFFFFFFull);
        g0.w = (unsigned int)((ga >> 32) & 0x1FFFFFFull) | (2u << 30);
        // D# group 1: data_size=4B, tensor 16384x1, tile 16384x1, stride 16384
        tdm_v8i g1;
        g1[0] = 0x00020000;                 // wg_mask=0, data_size=2 (4B)
        g1[1] = (int)(16384u << 16);        // tensor_dim0[15:0] at bits[63:48]
        g1[2] = (int)(1u << 16);            // tensor_dim0 hi=0 | tensor_dim1=1
        g1[3] = (int)(16384u << 16);        // tensor_dim1 hi=0 | tile_dim0=16384
        g1[4] = 1;                          // tile_dim1=1, tile_dim2 unused
        g1[5] = 16384;                      // tensor_dim0_stride lo32
        g1[6] = 0;                          // stride hi | dim1_stride lo
        g1[7] = 0;
        tdm_v4i g2 = {0, 0, 0, 0};
        tdm_v4i g3 = {0, 0, 0, 0};
        tdm_v8i g4 = {0, 0, 0, 0, 0, 0, 0, 0};  // 6-arg toolchain extra group
        __builtin_amdgcn_tensor_load_to_lds(g0, g1, g2, g3, g4, 0);
        __builtin_amdgcn_s_wait_tensorcnt(0);
    }
    __syncthreads();
#pragma unroll
    for (int i = 0; i < 16; ++i) {
        int g = i * 1024 + t;
        unsigned int kb = __float_as_uint(stage[g]);
        skey[g] = ((unsigned long long)kb << 32) | (unsigned int)g;
    }
#else
#pragma unroll
    for (int i = 0; i < 16; ++i) {
        int g = i * 1024 + t;
        unsigned int kb = __float_as_uint(dmin[(size_t)b * L_PTS + g]);
        skey[g] = ((unsigned long long)kb << 32) | (unsigned int)g;
    }
#endif
    __syncthreads();

    for (int k = 2; k <= L_PTS; k <<= 1) {
        for (int j = k >> 1; j > 0; j >>= 1) {
#pragma unroll 1
            for (int w = 0; w < 8; ++w) {
                int p = w * 1024 + t;                       // 0..8191 pair id
                int i = ((p & ~(j - 1)) << 1) | (p & (j - 1));
                int l = i | j;
                bool asc = ((i & k) == 0);
                unsigned long long va = skey[i], vb = skey[l];
                if ((va > vb) == asc) { skey[i] = vb; skey[l] = va; }
            }
            __syncthreads();
        }
    }

    for (int i = 0; i < 16; ++i) {
        int pos = i * 1024 + t;
        int idx = (int)(skey[pos] & 0xFFFFFFFFull);
        idxout[(size_t)b * L_PTS + pos] = idx;
        ids_restore[(size_t)b * L_PTS + idx] = (float)pos;
        mask[(size_t)b * L_PTS + idx] = (pos >= LEN_KEEP) ? 1.0f : 0.0f;
        const float* P = pts + ((size_t)b * L_PTS + idx) * 3;
        const float* R = rgb + ((size_t)b * L_PTS + idx) * 3;
        if (pos < LEN_KEEP) {
            float* dp = pkeep + ((size_t)b * LEN_KEEP + pos) * 3;
            float* dr = rkeep + ((size_t)b * LEN_KEEP + pos) * 3;
            dp[0] = P[0]; dp[1] = P[1]; dp[2] = P[2];
            dr[0] = R[0]; dr[1] = R[1]; dr[2] = R[2];
        } else {
            int q = pos - LEN_KEEP;
            float* dp = pmask + ((size_t)b * LEN_MASK + q) * 3;
            float* dr = rmask + ((size_t)b * LEN_MASK + q) * 3;
            dp[0] = P[0]; dp[1] = P[1]; dp[2] = P[2];
            dr[0] = R[0]; dr[1] = R[1]; dr[2] = R[2];
        }
    }
}

// ---------------------------------------------------------------------------
// Kernel 4: x_masked row gather, float4-vectorized (1 KB rows, 4 rows/block).
// Prefetch the gathered source row (global_prefetch_b8) before the loads.
// ---------------------------------------------------------------------------
__global__ void __launch_bounds__(256)
gather_x_kernel(const float* __restrict__ x, const int* __restrict__ idx,
                float* __restrict__ xout) {
    int row = blockIdx.x * 4 + (threadIdx.x >> 6);  // 0 .. N_B*LEN_KEEP-1
    int c4  = threadIdx.x & 63;                     // float4 column
    int b   = row >> 12;                            // / LEN_KEEP
    int pos = row & (LEN_KEEP - 1);
    int src = idx[(size_t)b * L_PTS + pos];
    const float4* S = (const float4*)(x + ((size_t)b * L_PTS + src) * D_DIM);
    __builtin_prefetch(S + c4, 0, 1);               // global_prefetch_b8
    float4* Dq = (float4*)(xout + (size_t)row * D_DIM);
    Dq[c4] = S[c4];
}

extern "C" void kernel_launch(void* const* d_in, const int* in_sizes, int n_in,
                              void* d_out, int out_size, void* d_ws, size_t ws_size,
                              hipStream_t stream) {
    const float* x     = (const float*)d_in[0];  // (8,16384,256)
    const float* pts   = (const float*)d_in[1];  // (8,16384,3)
    const float* rgb   = (const float*)d_in[2];  // (8,16384,3)
    const int*   finit = (const int*)d_in[3];    // (8,)

    // outputs, flat concat in reference return order
    float* out         = (float*)d_out;
    float* x_masked    = out;                          // 8*4096*256
    float* mask        = x_masked + (size_t)N_B * LEN_KEEP * D_DIM;  // 8*16384
    float* ids_restore = mask + (size_t)N_B * L_PTS;                 // 8*16384
    float* pkeep       = ids_restore + (size_t)N_B * L_PTS;          // 8*4096*3
    float* pmaskd      = pkeep + (size_t)N_B * LEN_KEEP * 3;         // 8*12288*3
    float* rkeep       = pmaskd + (size_t)N_B * LEN_MASK * 3;        // 8*4096*3
    float* rmaskd      = rkeep + (size_t)N_B * LEN_KEEP * 3;         // 8*12288*3

    // workspace
    float* wsf    = (float*)d_ws;
    float* cent   = wsf;                        // 8*25*3 floats (pad to 1024)
    float* dminws = wsf + 1024;                 // 8*16384 floats
    int*   idxws  = (int*)(wsf + 1024 + (size_t)N_B * L_PTS);  // 8*16384 ints

    fps_kernel<<<dim3(N_B), dim3(1024), 0, stream>>>(pts, finit, cent);
    dmin_kernel<<<dim3(L_PTS / 128, N_B), dim3(256), 0, stream>>>(pts, cent, dminws);
    sort_kernel<<<dim3(N_B), dim3(1024),
                  (size_t)L_PTS * sizeof(unsigned long long), stream>>>(
        dminws, pts, rgb, idxws, ids_restore, mask, pkeep, pmaskd, rkeep, rmaskd);
    gather_x_kernel<<<dim3((N_B * LEN_KEEP) / 4), dim3(256), 0, stream>>>(
        x, idxws, x_masked);
}